// Decoder_76407468196452
// MI455X (gfx1250) — compile-verified
//
#include <hip/hip_runtime.h>

// Problem constants (match the reference).
#define B_ 512
#define T_ 128
#define D_ 256
#define H_ 1024
#define KC 16           // K-chunk staged per LDS buffer
#define K1BUF 16384     // kernel-1 per-buffer LDS bytes (A 4KB + B 12KB)
#define K2BUF 4096      // kernel-2 per-buffer LDS bytes (A 2KB + B 2KB)

typedef __attribute__((ext_vector_type(2))) float v2f;
typedef __attribute__((ext_vector_type(8))) float v8f;

// D = A(16x4 f32) * B(4x16 f32) + C(16x16 f32), wave32.
__device__ __forceinline__ v8f wmma4(v2f a, v2f b, v8f c) {
    return __builtin_amdgcn_wmma_f32_16x16x4_f32(
        /*neg_a=*/false, a, /*neg_b=*/false, b,
        /*c_mod=*/(short)0, c, /*reuse_a=*/false, /*reuse_b=*/false);
}

// CDNA5 async global->LDS copy (8B per lane), tracked by ASYNCcnt.
__device__ __forceinline__ void async_copy_b64(unsigned lds_off, unsigned long long ga) {
    asm volatile("global_load_async_to_lds_b64 %0, %1, off"
                 :: "v"(lds_off), "v"(ga)
                 : "memory");
}

__device__ __forceinline__ void wait_async0() {
    asm volatile("s_wait_asynccnt 0x0" ::: "memory");
}

__device__ __forceinline__ float sigmoidf_(float x) {
    return 1.0f / (1.0f + __expf(-x));
}

// ---------------------------------------------------------------------------
// Kernel 1 helpers: 8 async copies per chunk with pointer-increment
// addressing; LDS offsets are loop-invariant per (call-site) buffer.
// ---------------------------------------------------------------------------
__device__ __forceinline__ void issue8(unsigned bufoff,
                                       const unsigned (&ldsA)[2],
                                       unsigned long long (&gaA)[2],
                                       const unsigned (&ldsB)[6],
                                       unsigned long long (&gaB)[6]) {
#pragma unroll
    for (int i = 0; i < 2; ++i) {
        async_copy_b64(ldsA[i] + bufoff, gaA[i]);
        gaA[i] += KC * 4;                 // advance 16 floats along K
    }
#pragma unroll
    for (int i = 0; i < 6; ++i) {
        async_copy_b64(ldsB[i] + bufoff, gaB[i]);
        gaB[i] += KC * 4;
    }
}

// One KC=16 chunk of WMMA work from LDS buffer at compile-time BUFOFF.
// acc_n is acc_xn in phase 0, acc_hn in phase 1 (chosen at call site).
template <int BUFOFF>
__device__ __forceinline__ void gate_compute(const char* smem, int arow, int col, int hi,
                                             v8f (&acc_r)[2], v8f (&acc_z)[2],
                                             v8f (&acc_n)[2]) {
    const v2f* As = (const v2f*)(smem + BUFOFF);           // [kk][64]
    const v2f* Bs = (const v2f*)(smem + BUFOFF + 4096);    // [g][kk][64]
#pragma unroll
    for (int ks = 0; ks < 4; ++ks) {
        const int kidx = ks * 2 + hi;
        const v2f br = Bs[(0 * 8 + kidx) * 64 + col];
        const v2f bz = Bs[(1 * 8 + kidx) * 64 + col];
        const v2f bn = Bs[(2 * 8 + kidx) * 64 + col];
#pragma unroll
        for (int ms = 0; ms < 2; ++ms) {
            const v2f a = As[kidx * 64 + arow + (ms << 4)];
            acc_r[ms] = wmma4(a, br, acc_r[ms]);
            acc_z[ms] = wmma4(a, bz, acc_z[ms]);
            acc_n[ms] = wmma4(a, bn, acc_n[ms]);
        }
    }
}

// ---------------------------------------------------------------------------
// Kernel 1: fused gates GEMM + GRU cell, async double-buffered LDS pipeline.
// Phase 0: 16 chunks of x*W_ih (K=D); phase 1: 64 chunks of h*W_hh (K=H).
// Block = 256 thr = 8 waves (2 x 4), tile 64(B-rows) x 64(H-cols).
// ---------------------------------------------------------------------------
__global__ __launch_bounds__(256)
void gru_gate_kernel(const float* __restrict__ x, int x_stride,
                     const float* __restrict__ hsrc,
                     const float* __restrict__ W_ih,
                     const float* __restrict__ W_hh,
                     const float* __restrict__ b_ih,
                     const float* __restrict__ b_hh,
                     float* __restrict__ hdst)
{
    extern __shared__ char smem[];          // 2 x K1BUF, base offset 0

    const int tid  = threadIdx.x;
    const int lane = tid & 31;
    const int wave = tid >> 5;
    const int wm   = wave >> 2;             // 0..1  (M half)
    const int wn   = wave & 3;              // 0..3  (N quarter)
    const int ln   = lane & 15;
    const int hi   = lane >> 4;             // K-pair select within fragment

    const int n0 = blockIdx.x * 64;         // column block within H
    const int m0 = blockIdx.y * 64;         // row block within B

    v8f acc_r[2]  = {};
    v8f acc_z[2]  = {};
    v8f acc_xn[2] = {};
    v8f acc_hn[2] = {};

    // ---- per-thread loader geometry (constant) ----
    // A: 512 v2f entries, 256 thr x 2; B: 1536 entries, 256 thr x 6.
    int aKK[2], aM[2], bG[6], bKK[6], bN[6];
    unsigned ldsA[2], ldsB[6];
#pragma unroll
    for (int i = 0; i < 2; ++i) {
        const int idx = tid + 256 * i;
        aKK[i] = idx >> 6;                  // 0..7
        aM[i]  = idx & 63;
        ldsA[i] = (unsigned)((aKK[i] * 64 + aM[i]) * 8);
    }
#pragma unroll
    for (int i = 0; i < 6; ++i) {
        const int idx = tid + 256 * i;
        bG[i]  = idx >> 9;                  // 0..2
        const int r = idx & 511;
        bKK[i] = r >> 6;                    // 0..7
        bN[i]  = r & 63;
        ldsB[i] = 4096u + (unsigned)(((bG[i] * 8 + bKK[i]) * 64 + bN[i]) * 8);
    }

    // ---- phase-0 global addresses (x, W_ih) ----
    unsigned long long gaA[2], gaB[6];
#pragma unroll
    for (int i = 0; i < 2; ++i)
        gaA[i] = (unsigned long long)(x + (size_t)(m0 + aM[i]) * x_stride + aKK[i] * 2);
#pragma unroll
    for (int i = 0; i < 6; ++i)
        gaB[i] = (unsigned long long)(W_ih + (size_t)(bG[i] * H_ + n0 + bN[i]) * D_
                                           + bKK[i] * 2);

    const int arow = (wm << 5) + ln;        // A fragment row base within tile
    const int col  = (wn << 4) + ln;        // B/N column within tile

    issue8(0, ldsA, gaA, ldsB, gaB);        // chunk 0 -> buf0

    // ---- phase 0: 16 chunks = 8 pairs ----
    for (int p = 0; p < 8; ++p) {
        wait_async0();
        __syncthreads();
        issue8(K1BUF, ldsA, gaA, ldsB, gaB);            // chunk 2p+1 -> buf1
        gate_compute<0>(smem, arow, col, hi, acc_r, acc_z, acc_xn);

        wait_async0();
        __syncthreads();
        if (p == 7) {
            // re-seed addresses for phase 1 (h, W_hh), then issue its chunk 0
#pragma unroll
            for (int i = 0; i < 2; ++i)
                gaA[i] = (unsigned long long)(hsrc + (size_t)(m0 + aM[i]) * H_
                                                   + aKK[i] * 2);
#pragma unroll
            for (int i = 0; i < 6; ++i)
                gaB[i] = (unsigned long long)(W_hh + (size_t)(bG[i] * H_ + n0 + bN[i]) * H_
                                                   + bKK[i] * 2);
        }
        issue8(0, ldsA, gaA, ldsB, gaB);                // chunk 2p+2 -> buf0
        gate_compute<K1BUF>(smem, arow, col, hi, acc_r, acc_z, acc_xn);
    }

    // ---- phase 1: 64 chunks = 32 pairs (chunk 0 already in flight) ----
    for (int p = 0; p < 32; ++p) {
        wait_async0();
        __syncthreads();
        issue8(K1BUF, ldsA, gaA, ldsB, gaB);            // chunk 2p+1 -> buf1
        gate_compute<0>(smem, arow, col, hi, acc_r, acc_z, acc_hn);

        wait_async0();
        __syncthreads();
        if (p < 31) issue8(0, ldsA, gaA, ldsB, gaB);    // chunk 2p+2 -> buf0
        gate_compute<K1BUF>(smem, arow, col, hi, acc_r, acc_z, acc_hn);
    }

    // ---- epilogue: GRU nonlinearity, h_new = (1-z)*n + z*h ----
    const int   j    = n0 + col;
    const float bb_r = b_ih[j] + b_hh[j];
    const float bb_z = b_ih[H_ + j] + b_hh[H_ + j];
    const float b_xn = b_ih[2 * H_ + j];
    const float b_hn = b_hh[2 * H_ + j];

#pragma unroll
    for (int ms = 0; ms < 2; ++ms) {
        const int rbase = m0 + (wm << 5) + (ms << 4) + (hi << 3);
#pragma unroll
        for (int v = 0; v < 8; ++v) {
            const int   row   = rbase + v;
            const float hprev = hsrc[(size_t)row * H_ + j];
            const float r = sigmoidf_(acc_r[ms][v] + bb_r);
            const float z = sigmoidf_(acc_z[ms][v] + bb_z);
            const float n = tanhf(acc_xn[ms][v] + b_xn + r * (acc_hn[ms][v] + b_hn));
            hdst[(size_t)row * H_ + j] = (1.0f - z) * n + z * hprev;
        }
    }
}

// ---------------------------------------------------------------------------
// Kernel 2: y[:,t,:] = x + h_new @ W_tp.T + b_tp, same pipeline scheme.
// Block = 128 thr = 4 waves (2 x 2), tile 32 x 32; wave = one 16x16 tile.
// ---------------------------------------------------------------------------
__device__ __forceinline__ void issue4(unsigned bufoff,
                                       const unsigned (&ldsA)[2],
                                       unsigned long long (&gaA)[2],
                                       const unsigned (&ldsB)[2],
                                       unsigned long long (&gaB)[2]) {
#pragma unroll
    for (int i = 0; i < 2; ++i) {
        async_copy_b64(ldsA[i] + bufoff, gaA[i]);
        gaA[i] += KC * 4;
    }
#pragma unroll
    for (int i = 0; i < 2; ++i) {
        async_copy_b64(ldsB[i] + bufoff, gaB[i]);
        gaB[i] += KC * 4;
    }
}

template <int BUFOFF>
__device__ __forceinline__ void proj_compute(const char* smem, int arow, int col, int hi,
                                             v8f& acc) {
    const v2f* As = (const v2f*)(smem + BUFOFF);           // [kk][32]
    const v2f* Bs = (const v2f*)(smem + BUFOFF + 2048);    // [kk][32]
#pragma unroll
    for (int ks = 0; ks < 4; ++ks) {
        const int kidx = ks * 2 + hi;
        const v2f b = Bs[kidx * 32 + col];
        const v2f a = As[kidx * 32 + arow];
        acc = wmma4(a, b, acc);
    }
}

__global__ __launch_bounds__(128)
void proj_kernel(const float* __restrict__ hnew,
                 const float* __restrict__ W_tp,
                 const float* __restrict__ b_tp,
                 const float* __restrict__ xprev,  // row stride T_*D_
                 float* __restrict__ yout)         // row stride T_*D_ (pre-offset t*D_)
{
    extern __shared__ char smem[];          // 2 x K2BUF

    const int tid  = threadIdx.x;
    const int lane = tid & 31;
    const int wave = tid >> 5;
    const int wm   = wave >> 1;
    const int wn   = wave & 1;
    const int ln   = lane & 15;
    const int hi   = lane >> 4;

    const int n0 = blockIdx.x * 32;         // column block within D
    const int m0 = blockIdx.y * 32;         // row block within B

    v8f acc = {};

    // loader geometry: A 256 v2f / 128 thr x 2; B likewise.
    unsigned ldsA[2], ldsB[2];
    unsigned long long gaA[2], gaB[2];
#pragma unroll
    for (int i = 0; i < 2; ++i) {
        const int idx = tid + 128 * i;
        const int kk  = idx >> 5;           // 0..7
        const int m   = idx & 31;
        ldsA[i] = (unsigned)((kk * 32 + m) * 8);
        ldsB[i] = 2048u + ldsA[i] - (unsigned)((kk * 32 + m) * 8)
                  + (unsigned)((kk * 32 + m) * 8);   // = 2048 + same pattern
        ldsB[i] = 2048u + (unsigned)((kk * 32 + m) * 8);
        gaA[i]  = (unsigned long long)(hnew + (size_t)(m0 + m) * H_ + kk * 2);
        gaB[i]  = (unsigned long long)(W_tp + (size_t)(n0 + m) * H_ + kk * 2);
    }

    const int arow = (wm << 4) + ln;
    const int col  = (wn << 4) + ln;

    issue4(0, ldsA, gaA, ldsB, gaB);        // chunk 0 -> buf0

    // H_/KC = 64 chunks = 32 pairs
    for (int p = 0; p < 32; ++p) {
        wait_async0();
        __syncthreads();
        issue4(K2BUF, ldsA, gaA, ldsB, gaB);            // chunk 2p+1 -> buf1
        proj_compute<0>(smem, arow, col, hi, acc);

        wait_async0();
        __syncthreads();
        if (p < 31) issue4(0, ldsA, gaA, ldsB, gaB);    // chunk 2p+2 -> buf0
        proj_compute<K2BUF>(smem, arow, col, hi, acc);
    }

    const int   j     = n0 + col;
    const float bt    = b_tp[j];
    const int   rbase = m0 + (wm << 4) + (hi << 3);
#pragma unroll
    for (int v = 0; v < 8; ++v) {
        const int row = rbase + v;
        const float xv = xprev[(size_t)row * (T_ * D_) + j];
        yout[(size_t)row * (T_ * D_) + j] = acc[v] + bt + xv;
    }
}

// ---------------------------------------------------------------------------
// Host: 128 sequential steps, 2 kernels per step (graph-captured).
// h ping-pongs through d_ws (2 x 2 MB); y chains through d_out directly.
// ---------------------------------------------------------------------------
extern "C" void kernel_launch(void* const* d_in, const int* in_sizes, int n_in,
                              void* d_out, int out_size, void* d_ws, size_t ws_size,
                              hipStream_t stream) {
    (void)in_sizes; (void)n_in; (void)out_size; (void)ws_size;

    const float* h0   = (const float*)d_in[0];
    const float* gt   = (const float*)d_in[1];
    const float* W_ih = (const float*)d_in[2];
    const float* W_hh = (const float*)d_in[3];
    const float* b_ih = (const float*)d_in[4];
    const float* b_hh = (const float*)d_in[5];
    const float* W_tp = (const float*)d_in[6];
    const float* b_tp = (const float*)d_in[7];

    float* out = (float*)d_out;
    float* hA  = (float*)d_ws;
    float* hB  = hA + (size_t)B_ * H_;

    const dim3 g1(H_ / 64, B_ / 64);   // 16 x 8 blocks, 256 thr, 32KB LDS
    const dim3 g2(D_ / 32, B_ / 32);   // 8 x 16 blocks, 128 thr, 8KB LDS

    for (int t = 0; t < T_; ++t) {
        // x_t: gt[:,0,:] for t==0, else y_{t-1} living inside d_out.
        // Both have row stride T_*D_.
        const float* x  = (t == 0) ? gt : (out + (size_t)(t - 1) * D_);
        const float* hs = (t == 0) ? h0 : ((t & 1) ? hA : hB);
        float*       hd = (t & 1) ? hB : hA;

        gru_gate_kernel<<<g1, 256, 2 * K1BUF, stream>>>(x, T_ * D_, hs,
                                                        W_ih, W_hh, b_ih, b_hh, hd);
        proj_kernel<<<g2, 128, 2 * K2BUF, stream>>>(hd, W_tp, b_tp, x,
                                                    out + (size_t)t * D_);
    }
}